// DMGI_13975823582294
// MI455X (gfx1250) — compile-verified
//
#include <hip/hip_runtime.h>

#define NN   100000   // num_nodes
#define RR   3        // num_relations
#define EE   1600000  // edges per relation
#define DIN  128
#define DOUT 64

typedef __attribute__((ext_vector_type(2))) float v2f;
typedef __attribute__((ext_vector_type(8))) float v8f;

// ---------------------------------------------------------------- utilities
__global__ void fill_kernel(float* __restrict__ p, long long n, float v) {
    long long i = (long long)blockIdx.x * blockDim.x + threadIdx.x;
    long long stride = (long long)gridDim.x * blockDim.x;
    for (; i < n; i += stride) p[i] = v;
}

__global__ void scale_kernel(float* __restrict__ p, int n, float s) {
    int i = blockIdx.x * blockDim.x + threadIdx.x;
    if (i < n) p[i] *= s;
}

// ------------------------------------------------------ degree / normalization
__global__ void deg_kernel(const int* __restrict__ dst, float* __restrict__ deg, int nE) {
    int e = blockIdx.x * blockDim.x + threadIdx.x;
    if (e < nE) atomicAdd(&deg[dst[e]], 1.0f);
}

__global__ void dinv_kernel(float* __restrict__ deg, int n) {
    int i = blockIdx.x * blockDim.x + threadIdx.x;
    if (i < n) deg[i] = rsqrtf(deg[i] + 1.0f);  // +1 = self loop; always > 0
}

// ---------------------------------------------------------------- WMMA GEMM
// h[NN x 64] = x[NN x 128] @ W[128 x 64], one wave per 16x16 output tile.
// V_WMMA_F32_16X16X4_F32 lane layouts (ISA 7.12.2):
//   A (16x4):  lane m = lane&15 holds row m; K = vgpr + 2*(lane>=16)
//   B (4x16):  lane n = lane&15 holds col n; K = vgpr + 2*(lane>=16)
//   C/D (16x16): lane n = lane&15; M = vgpr + 8*(lane>=16)
__global__ void gemm_wmma_kernel(const float* __restrict__ x,
                                 const float* __restrict__ W,
                                 float* __restrict__ h) {
    const int waveId = blockIdx.x * (blockDim.x >> 5) + (threadIdx.x >> 5);
    const int lane   = threadIdx.x & 31;
    const int rowTile = waveId >> 2;     // 0 .. 6249
    const int colTile = waveId & 3;      // 0 .. 3
    if (rowTile >= NN / 16) return;      // wave-uniform (EXEC stays all-ones)

    const int row0 = rowTile << 4;
    const int col0 = colTile << 4;
    const int m  = lane & 15;
    const int hi = lane >> 4;            // 0 or 1 -> K offset 0 or 2

    const float* xrow = x + (long long)(row0 + m) * DIN + 2 * hi;
    const float* wcol = W + (long long)(2 * hi) * DOUT + col0 + m;

    v8f acc = {};
#pragma unroll
    for (int k0 = 0; k0 < DIN; k0 += 4) {
        v2f a, b;
        a.x = xrow[k0];                              // x[row][k0 + 2*hi]
        a.y = xrow[k0 + 1];                          // x[row][k0 + 2*hi + 1]
        b.x = wcol[(long long)k0 * DOUT];            // W[k0 + 2*hi][col]
        b.y = wcol[(long long)k0 * DOUT + DOUT];     // W[k0 + 2*hi + 1][col]
        acc = __builtin_amdgcn_wmma_f32_16x16x4_f32(
            /*neg_a=*/false, a, /*neg_b=*/false, b,
            /*c_mod=*/(short)0, acc, /*reuse_a=*/false, /*reuse_b=*/false);
    }

    float* hrow = h + (long long)(row0 + 8 * hi) * DOUT + col0 + m;
#pragma unroll
    for (int i = 0; i < 8; ++i)
        hrow[(long long)i * DOUT] = acc[i];
}

// ------------------------------------------------------------ edge scatter
// One wave per edge. Lane owns columns {2*lane, 2*lane+1}:
// coalesced 256B gather of h[src] per wave, then 64 global f32 atomics -> L2.
__global__ void scatter_kernel(const float* __restrict__ h,
                               const float* __restrict__ dinv,
                               const int* __restrict__ src,
                               const int* __restrict__ dst,
                               const int* __restrict__ perm, const int use_perm,
                               float* __restrict__ out, const int nE) {
    const int wave = blockIdx.x * (blockDim.x >> 5) + (threadIdx.x >> 5);
    const int lane = threadIdx.x & 31;
    if (wave >= nE) return;

    const int s = src[wave];
    const int d = dst[wave];
    const float w = dinv[s] * dinv[d];
    const int hs = use_perm ? perm[s] : s;

    const float2 v = *(const float2*)(h + (long long)hs * DOUT + (lane << 1));
    float* o = out + (long long)d * DOUT + (lane << 1);
    atomicAdd(o,     v.x * w);
    atomicAdd(o + 1, v.y * w);
}

// ---------------------------------------------------------------- finalize
// out = relu(agg + h[perm?][i]*dinv[i]^2 + bias); optional column-sum for summary
// (LDS partial sums, one global atomic per column per block).
__global__ void finalize_kernel(float* __restrict__ out,
                                const float* __restrict__ h,
                                const float* __restrict__ dinv,
                                const float* __restrict__ bias,
                                const int* __restrict__ perm, const int use_perm,
                                float* __restrict__ summ, const int do_summary) {
    __shared__ float sdata[DOUT];
    const int tid = threadIdx.x;
    if (tid < DOUT) sdata[tid] = 0.0f;
    __syncthreads();

    const long long idx = (long long)blockIdx.x * blockDim.x + tid;
    const int node = (int)(idx >> 6);   // / DOUT
    const int c    = (int)(idx & 63);   // % DOUT

    float val = 0.0f;
    if (node < NN) {
        const float di = dinv[node];
        const int hs = use_perm ? perm[node] : node;
        val = out[idx] + h[(long long)hs * DOUT + c] * (di * di) + bias[c];
        val = fmaxf(val, 0.0f);
        out[idx] = val;
    }

    if (do_summary) {
        atomicAdd(&sdata[c], val);      // ds_add_f32
        __syncthreads();
        if (tid < DOUT) atomicAdd(&summ[tid], sdata[tid]);
    }
}

// ---------------------------------------------------------------- launcher
extern "C" void kernel_launch(void* const* d_in, const int* in_sizes, int n_in,
                              void* d_out, int out_size, void* d_ws, size_t ws_size,
                              hipStream_t stream) {
    const float* x     = (const float*)d_in[0];   // [NN, 128]
    const float* W     = (const float*)d_in[1];   // [RR, 128, 64]
    const float* bias  = (const float*)d_in[2];   // [RR, 64]
    const int*   ei    = (const int*)d_in[3];     // [RR, 2, EE]
    const int*   perms = (const int*)d_in[4];     // [RR, NN]

    float* pos  = (float*)d_out;                          // [RR, NN, 64]
    float* neg  = pos + (long long)RR * NN * DOUT;        // [RR, NN, 64]
    float* summ = neg + (long long)RR * NN * DOUT;        // [RR, 64]

    // workspace: h (NN*64 f32, 25.6 MB) + deg/dinv (NN f32)
    float* h   = (float*)d_ws;
    float* deg = h + (long long)NN * DOUT;

    fill_kernel<<<64, 256, 0, stream>>>(summ, RR * DOUT, 0.0f);

    for (int r = 0; r < RR; ++r) {
        const int* src    = ei + (long long)r * 2 * EE;
        const int* dst    = src + EE;
        const int* perm   = perms + (long long)r * NN;
        float* pos_r      = pos + (long long)r * NN * DOUT;
        float* neg_r      = neg + (long long)r * NN * DOUT;
        float* summ_r     = summ + r * DOUT;
        const float* W_r  = W + (long long)r * DIN * DOUT;
        const float* b_r  = bias + r * DOUT;

        // symmetric normalization
        fill_kernel<<<256, 256, 0, stream>>>(deg, NN, 0.0f);
        deg_kernel<<<(EE + 255) / 256, 256, 0, stream>>>(dst, deg, EE);
        dinv_kernel<<<(NN + 255) / 256, 256, 0, stream>>>(deg, NN);

        // h = x @ W_r  (6250*4 = 25000 wave-tiles, 8 waves/block)
        gemm_wmma_kernel<<<(NN / 16) * 4 / 8, 256, 0, stream>>>(x, W_r, h);

        // accumulate into outputs
        fill_kernel<<<1024, 256, 0, stream>>>(pos_r, (long long)NN * DOUT, 0.0f);
        fill_kernel<<<1024, 256, 0, stream>>>(neg_r, (long long)NN * DOUT, 0.0f);

        scatter_kernel<<<(EE + 7) / 8, 256, 0, stream>>>(h, deg, src, dst, perm, 0, pos_r, EE);
        scatter_kernel<<<(EE + 7) / 8, 256, 0, stream>>>(h, deg, src, dst, perm, 1, neg_r, EE);

        // self-loop + bias + relu (+ summary column-sum for pos)
        finalize_kernel<<<(NN * DOUT) / 256, 256, 0, stream>>>(pos_r, h, deg, b_r, perm, 0, summ_r, 1);
        finalize_kernel<<<(NN * DOUT) / 256, 256, 0, stream>>>(neg_r, h, deg, b_r, perm, 1, summ_r, 0);
    }

    // summary = column mean
    scale_kernel<<<1, 256, 0, stream>>>(summ, RR * DOUT, 1.0f / (float)NN);
}